// RevIN_weight_27453430956433
// MI455X (gfx1250) — compile-verified
//
#include <hip/hip_runtime.h>
#include <math.h>

#define BB 256
#define TT 2048
#define CC 32
#define KTOP 5
#define BC (BB*CC)                 // 8192 per-(b,c) scalars per array
#define BTC ((size_t)BB*(size_t)TT*(size_t)CC)

// ws float layout (22*BC floats = 720,896 B):
//  0*BC conv_tm   1*BC conv_pm   2*BC conv_rlx  3*BC conv_rlt  4*BC conv_rlp
//  5*BC multi_trend   6*BC ratio
//  7*BC..12*BC  multi_p[j][BC]
// 12*BC..17*BC  p_f[j][BC]   (rounded, conditioned period as float)
// 17*BC..22*BC  rate[j][BC]

typedef float v2f __attribute__((ext_vector_type(2)));
typedef float v8f __attribute__((ext_vector_type(8)));

__device__ __forceinline__ v8f wmma_f32_k4(v2f a, v2f b, v8f c) {
  // D = A(16x4 f32) * B(4x16 f32) + C(16x16 f32)
  return __builtin_amdgcn_wmma_f32_16x16x4_f32(false, a, false, b, (short)0, c, false, false);
}

// ---------------------------------------------------------------------------
// K1: per-b column reductions: L2 norms + time-reduced conv sums,
// then circular channel mix -> conv_tm, conv_pm, conv_rlx  [B,C]
// ---------------------------------------------------------------------------
__global__ __launch_bounds__(256) void k1_colred(
    const float* __restrict__ x, const float* __restrict__ xi,
    const float* __restrict__ tm_conv, const float* __restrict__ pm_conv,
    const float* __restrict__ rl_convx, float* __restrict__ ws)
{
  const int b = blockIdx.x, c = threadIdx.x, tg = threadIdx.y;
  const float* xb = x + (size_t)b*TT*CC;
  const float* ib = xi + (size_t)b*TT*CC;
  float acc[11];
#pragma unroll
  for (int i = 0; i < 11; ++i) acc[i] = 0.f;
  for (int t = tg; t < TT; t += 8) {
    float xv = xb[(size_t)t*CC + c];
    float iv = ib[(size_t)t*CC + c];
    acc[0] += xv*xv;
    acc[1] += iv*iv;
#pragma unroll
    for (int k = 0; k < 3; ++k) {
      acc[2+k] += xv * tm_conv[t*3+k];
      acc[5+k] += iv * pm_conv[t*3+k];
      acc[8+k] += iv * rl_convx[t*3+k];
    }
  }
  __shared__ float red[11][8][32];
  __shared__ float fin[9][32];
#pragma unroll
  for (int i = 0; i < 11; ++i) red[i][tg][c] = acc[i];
  __syncthreads();
  if (tg == 0) {
    float s[11];
#pragma unroll
    for (int i = 0; i < 11; ++i) {
      float v = 0.f;
#pragma unroll
      for (int g = 0; g < 8; ++g) v += red[i][g][c];
      s[i] = v;
    }
    float nx = fmaxf(sqrtf(s[0]), 1e-12f);
    float ni = fmaxf(sqrtf(s[1]), 1e-12f);
#pragma unroll
    for (int k = 0; k < 3; ++k) {
      fin[k][c]   = s[2+k] / nx;   // x_d . tm_conv[:,k]
      fin[3+k][c] = s[5+k] / ni;   // x_int . pm_conv[:,k]
      fin[6+k][c] = s[8+k] / ni;   // x_int . rl_convx[:,k]
    }
  }
  __syncthreads();
  if (tg == 0) {
    float ctm = 0.f, cpm = 0.f, crx = 0.f;
#pragma unroll
    for (int k = 0; k < 3; ++k) {
      int c2 = (c + k + CC - 1) & (CC - 1);     // circular pad along channels
      ctm += fin[k][c2];
      cpm += fin[3+k][c2];
      crx += fin[6+k][c2];
    }
    ws[0*BC + b*CC + c] = ctm;
    ws[1*BC + b*CC + c] = cpm;
    ws[2*BC + b*CC + c] = crx;
  }
}

// ---------------------------------------------------------------------------
// K2: per-(b,c) 2048-pt Stockham radix-2 FFT in LDS + |X| top-5 -> p_f, rate
// ---------------------------------------------------------------------------
__global__ __launch_bounds__(128) void k2_fft_topk(
    const float* __restrict__ xi, float* __restrict__ ws)
{
  const int bc = blockIdx.x;
  const int b = bc >> 5, c = bc & 31;
  const int tid = threadIdx.x;
  __shared__ float2 buf[2][TT];
  __shared__ float amp[TT/2];
  __shared__ float rv[128];
  __shared__ int   ri[128];
  __shared__ float pv[KTOP];
  __shared__ int   pidx[KTOP];

  const float* col = xi + (size_t)b*TT*CC + c;
  for (int u = tid; u < TT; u += 128)
    buf[0][u] = make_float2(col[(size_t)u*CC], 0.f);
  __syncthreads();

  int src = 0, s = 1, ls = 0, ncur = TT;
  while (ncur > 1) {
    const int m = ncur >> 1;
    const float th0 = -6.283185307179586f / (float)ncur;
    for (int u = tid; u < TT/2; u += 128) {
      const int q = u & (s - 1);
      const int p = u >> ls;
      float2 a  = buf[src][q + s*p];
      float2 bv = buf[src][q + s*(p + m)];
      float ang = th0 * (float)p;
      float wr = cosf(ang), wi = sinf(ang);
      float dx = a.x - bv.x, dy = a.y - bv.y;
      buf[src^1][q + 2*s*p]     = make_float2(a.x + bv.x, a.y + bv.y);
      buf[src^1][q + 2*s*p + s] = make_float2(dx*wr - dy*wi, dx*wi + dy*wr);
    }
    __syncthreads();
    src ^= 1; s <<= 1; ++ls; ncur >>= 1;
  }
  for (int u = tid; u < TT/2; u += 128) {
    float2 v = buf[src][u];
    amp[u] = sqrtf(v.x*v.x + v.y*v.y);
  }
  __syncthreads();

  for (int it = 0; it < KTOP; ++it) {
    float bvv = -1.f; int bii = 1 << 30;
    for (int u = tid; u < TT/2; u += 128) {
      float v = amp[u];
      if (v > bvv || (v == bvv && u < bii)) { bvv = v; bii = u; }
    }
    rv[tid] = bvv; ri[tid] = bii;
    __syncthreads();
    for (int off = 64; off > 0; off >>= 1) {
      if (tid < off) {
        float v2 = rv[tid + off]; int i2 = ri[tid + off];
        if (v2 > rv[tid] || (v2 == rv[tid] && i2 < ri[tid])) { rv[tid] = v2; ri[tid] = i2; }
      }
      __syncthreads();
    }
    if (tid == 0) { pv[it] = rv[0]; pidx[it] = ri[0]; amp[ri[0]] = -1.f; }
    __syncthreads();
  }
  if (tid == 0) {
    float asum = 1e-5f;               // EPS
    for (int j = 0; j < KTOP; ++j) asum += pv[j];
    for (int j = 0; j < KTOP; ++j) {
      int id = pidx[j];
      // p = T/idx; keep iff finite and < T  (idx>=2), else 1.0; round half-even
      float pf = (id >= 2) ? rintf((float)TT / (float)id) : 1.0f;
      ws[12*BC + j*BC + b*CC + c] = pf;
      ws[17*BC + j*BC + b*CC + c] = pv[j] / asum;
    }
  }
}

// ---------------------------------------------------------------------------
// WMMA MLP on an LDS-staged Z tile (16 x KDIM, row stride ZS). Branch-free.
// A-layout (16x4 f32): lane m=lane&15, half=lane>>4 -> K = kk*4 + 2*half + r
// B-layout mirrors it; C/D rows M = r + 8*half, cols N = lane&15.
// ---------------------------------------------------------------------------
template <int KDIM, int ZS>
__device__ __forceinline__ void mlp_fwd(const float* z,
    const float* __restrict__ W1, const float* __restrict__ B1,
    const float* __restrict__ W2, const float* __restrict__ B2,
    const float* __restrict__ W3, float out[2][8])
{
  const int lane = threadIdx.x;
  const int m = lane & 15;
  const int half = lane >> 4;
  __shared__ float h1[16*17];
  __shared__ float h2[16*17];

  v8f acc;
#pragma unroll
  for (int i = 0; i < 8; ++i) acc[i] = 0.f;
#pragma unroll
  for (int kk = 0; kk < KDIM/4; ++kk) {
    const int k0 = kk*4 + half*2;
    v2f a, bm;
    a.x = z[m*ZS + k0];
    a.y = z[m*ZS + k0 + 1];
    bm.x = W1[m*KDIM + k0];
    bm.y = W1[m*KDIM + k0 + 1];
    acc = wmma_f32_k4(a, bm, acc);
  }
#pragma unroll
  for (int r = 0; r < 8; ++r)
    h1[(r + 8*half)*17 + m] = fmaxf(acc[r] + B1[m], 0.f);
  __syncthreads();

#pragma unroll
  for (int i = 0; i < 8; ++i) acc[i] = 0.f;
#pragma unroll
  for (int kk = 0; kk < 4; ++kk) {
    const int k0 = kk*4 + half*2;
    v2f a, bm;
    a.x = h1[m*17 + k0];  a.y = h1[m*17 + k0 + 1];
    bm.x = W2[m*16 + k0]; bm.y = W2[m*16 + k0 + 1];
    acc = wmma_f32_k4(a, bm, acc);
  }
#pragma unroll
  for (int r = 0; r < 8; ++r)
    h2[(r + 8*half)*17 + m] = fmaxf(acc[r] + B2[m], 0.f);
  __syncthreads();

#pragma unroll
  for (int nt = 0; nt < 2; ++nt) {
    v8f a3;
#pragma unroll
    for (int i = 0; i < 8; ++i) a3[i] = 0.f;
#pragma unroll
    for (int kk = 0; kk < 4; ++kk) {
      const int k0 = kk*4 + half*2;
      v2f a, bm;
      a.x = h2[m*17 + k0]; a.y = h2[m*17 + k0 + 1];
      bm.x = W3[(nt*16 + m)*16 + k0];
      bm.y = W3[(nt*16 + m)*16 + k0 + 1];
      a3 = wmma_f32_k4(a, bm, a3);
    }
#pragma unroll
    for (int r = 0; r < 8; ++r) out[nt][r] = a3[r];
  }
}

// K3: prob 0 = trend MLP; prob 1..5 = periodic MLP j  -> exp(tanh(.))
__global__ __launch_bounds__(32) void k3_mlp(
    float* __restrict__ ws,
    const float* __restrict__ tm_w1, const float* __restrict__ tm_b1,
    const float* __restrict__ tm_w2, const float* __restrict__ tm_b2,
    const float* __restrict__ tm_w3,
    const float* __restrict__ pm_w1, const float* __restrict__ pm_b1,
    const float* __restrict__ pm_w2, const float* __restrict__ pm_b2,
    const float* __restrict__ pm_w3)
{
  const int lane = threadIdx.x;
  const int mtile = blockIdx.x;          // 16 tiles of 16 batch rows
  const int prob = blockIdx.y;           // 0..5 (wave-uniform)
  const int rowbase = mtile * 16;
  const float* W1 = prob ? pm_w1 : tm_w1;
  const float* B1 = prob ? pm_b1 : tm_b1;
  const float* W2 = prob ? pm_w2 : tm_w2;
  const float* B2 = prob ? pm_b2 : tm_b2;
  const float* W3 = prob ? pm_w3 : tm_w3;

  // Stage Z = [conv, ones|p_f] into LDS with uniform, coalesced loads.
  __shared__ float zsh[16*65];
  if (prob == 0) {
#pragma unroll
    for (int r = 0; r < 16; ++r) {
      zsh[r*65 + lane]      = ws[0*BC + (rowbase + r)*CC + lane];
      zsh[r*65 + 32 + lane] = 1.0f;
    }
  } else {
    const int j = prob - 1;
#pragma unroll
    for (int r = 0; r < 16; ++r) {
      zsh[r*65 + lane]      = ws[1*BC + (rowbase + r)*CC + lane];
      zsh[r*65 + 32 + lane] = ws[12*BC + j*BC + (rowbase + r)*CC + lane];
    }
  }
  __syncthreads();

  float out[2][8];
  mlp_fwd<64, 65>(zsh, W1, B1, W2, B2, W3, out);

  const int n = lane & 15, half = lane >> 4;
  float* dst = ws + (prob ? (7*BC + (size_t)(prob - 1)*BC) : (size_t)(5*BC));
#pragma unroll
  for (int nt = 0; nt < 2; ++nt)
#pragma unroll
    for (int r = 0; r < 8; ++r) {
      int row = rowbase + r + 8*half;
      dst[row*CC + nt*16 + n] = expf(tanhf(out[nt][r]));
    }
}

// ---------------------------------------------------------------------------
// K4: build weight_trend / weight_period from shifted mask (LDS bitmask),
// stage them into d_out, fuse rl_convt / rl_convp reductions.
// ---------------------------------------------------------------------------
__global__ __launch_bounds__(256) void k4_build_weights(
    const int* __restrict__ mask,
    const float* __restrict__ rl_convt, const float* __restrict__ rl_convp,
    float* __restrict__ ws, float* __restrict__ out)
{
  const int b = blockIdx.x, c = threadIdx.x, tg = threadIdx.y;
  __shared__ unsigned mbits[CC*64];            // bit (t%32) of word c*64 + t/32
  __shared__ float red[6][8][32];
  __shared__ float fin[6][32];
  const int* mb = mask + (size_t)b*TT*CC;
  for (int w = tg; w < 64; w += 8) {
    unsigned bits = 0;
    const int tbase = w * 32;
    for (int jj = 0; jj < 32; ++jj)
      bits |= (unsigned)(mb[(size_t)(tbase + jj)*CC + c] == 0) << jj;
    mbits[c*64 + w] = bits;
  }
  __syncthreads();

  const int bc = b*CC + c;
  const float mt = ws[5*BC + bc];
  float mp[KTOP], rt[KTOP]; int pi[KTOP];
#pragma unroll
  for (int j = 0; j < KTOP; ++j) {
    mp[j] = ws[7*BC  + j*BC + bc];
    rt[j] = ws[17*BC + j*BC + bc];
    pi[j] = (int)ws[12*BC + j*BC + bc];
  }
  const unsigned* mrow = mbits + c*64;
  auto M0 = [&](int t) -> float {
    if (t < 0 || t >= TT) return 0.f;
    return (float)((mrow[t >> 5] >> (t & 31)) & 1u);
  };
  const float g1 = 0.24197072451914337f;   // N(i;sigma=1), i=1
  const float g2 = 0.05399096651318806f;   // i=2
  const float g3 = 0.004431848411938008f;  // i=3
  float st[3] = {0,0,0}, sp[3] = {0,0,0};
  float* wt_out = out + (size_t)b*TT*CC;         // xn slot (staging)
  float* wp_out = out + BTC + (size_t)b*TT*CC;   // weight slot (staging)

  for (int t = tg; t < TT; t += 8) {
    float g = g1*(M0(t-1)+M0(t+1)) + g2*(M0(t-2)+M0(t+2)) + g3*(M0(t-3)+M0(t+3));
    float wt = 1.f + mt*g;
    float wp = 0.f;
#pragma unroll
    for (int j = 0; j < KTOP; ++j) {
      int p = pi[j];
      float gj = g1*(M0(t-p)+M0(t+p)) + g2*(M0(t-2*p)+M0(t+2*p)) + g3*(M0(t-3*p)+M0(t+3*p));
      wp += rt[j]*(1.f + mp[j]*gj);
    }
    wt_out[(size_t)t*CC + c] = wt;
    wp_out[(size_t)t*CC + c] = wp;
#pragma unroll
    for (int k = 0; k < 3; ++k) {
      st[k] += wt * rl_convt[t*3+k];
      sp[k] += wp * rl_convp[t*3+k];
    }
  }
#pragma unroll
  for (int k = 0; k < 3; ++k) { red[k][tg][c] = st[k]; red[3+k][tg][c] = sp[k]; }
  __syncthreads();
  if (tg == 0) {
#pragma unroll
    for (int i = 0; i < 6; ++i) {
      float v = 0.f;
#pragma unroll
      for (int g = 0; g < 8; ++g) v += red[i][g][c];
      fin[i][c] = v;
    }
  }
  __syncthreads();
  if (tg == 0) {
    float crt = 0.f, crp = 0.f;
#pragma unroll
    for (int k = 0; k < 3; ++k) {
      int c2 = (c + k + CC - 1) & (CC - 1);
      crt += fin[k][c2];
      crp += fin[3+k][c2];
    }
    ws[3*BC + bc] = crt;
    ws[4*BC + bc] = crp;
  }
}

// K5: ratio MLP (K=96, zr = [conv_rlx, conv_rlt, conv_rlp]) -> sigmoid
__global__ __launch_bounds__(32) void k5_ratio(
    float* __restrict__ ws,
    const float* __restrict__ rl_w1, const float* __restrict__ rl_b1,
    const float* __restrict__ rl_w2, const float* __restrict__ rl_b2,
    const float* __restrict__ rl_w3)
{
  const int lane = threadIdx.x;
  const int mtile = blockIdx.x;
  const int rowbase = mtile * 16;

  __shared__ float zsh[16*97];
#pragma unroll
  for (int seg = 0; seg < 3; ++seg)
#pragma unroll
    for (int r = 0; r < 16; ++r)
      zsh[r*97 + seg*32 + lane] = ws[(2 + seg)*BC + (rowbase + r)*CC + lane];
  __syncthreads();

  float out[2][8];
  mlp_fwd<96, 97>(zsh, rl_w1, rl_b1, rl_w2, rl_b2, rl_w3, out);

  const int n = lane & 15, half = lane >> 4;
#pragma unroll
  for (int nt = 0; nt < 2; ++nt)
#pragma unroll
    for (int r = 0; r < 8; ++r) {
      int row = rowbase + r + 8*half;
      float v = out[nt][r];
      ws[6*BC + row*CC + nt*16 + n] = 1.f / (1.f + expf(-v));
    }
}

// ---------------------------------------------------------------------------
// K6: blend + masked weighted normalization; overwrites staging in d_out.
// ---------------------------------------------------------------------------
__global__ __launch_bounds__(256) void k6_final(
    const float* __restrict__ x, const int* __restrict__ mask,
    const float* __restrict__ ws, float* __restrict__ out)
{
  const int b = blockIdx.x, c = threadIdx.x, tg = threadIdx.y;
  const int bc = b*CC + c;
  const float ratio = ws[6*BC + bc];
  const float* xb = x + (size_t)b*TT*CC;
  const int* mb = mask + (size_t)b*TT*CC;
  float* xn_out = out + (size_t)b*TT*CC;         // holds staged wt on entry
  float* w_out  = out + BTC + (size_t)b*TT*CC;   // holds staged wp on entry
  __shared__ float red[2][8][32];
  __shared__ float cntsh[32], meansh[32], stdsh[32];

  float cw = 0.f, cxw = 0.f;
  for (int t = tg; t < TT; t += 8) {
    size_t idx = (size_t)t*CC + c;
    float wtv = xn_out[idx];
    float wpv = w_out[idx];
    float mf = (mb[idx] != 0) ? 1.f : 0.f;
    float w = (wtv*ratio + wpv*(1.f - ratio)) * mf;
    w_out[idx] = w;                 // final weight output (weight*maskf)
    cw  += w;
    cxw += xb[idx] * w;
  }
  red[0][tg][c] = cw; red[1][tg][c] = cxw;
  __syncthreads();
  if (tg == 0) {
    float s0 = 0.f, s1 = 0.f;
#pragma unroll
    for (int g = 0; g < 8; ++g) { s0 += red[0][g][c]; s1 += red[1][g][c]; }
    float cnt = (s0 == 0.f) ? 1.f : s0;
    cntsh[c] = cnt;
    meansh[c] = s1 / cnt;
  }
  __syncthreads();
  const float mean = meansh[c], cnt = cntsh[c];
  float acc = 0.f;
  for (int t = tg; t < TT; t += 8) {
    size_t idx = (size_t)t*CC + c;
    float w = w_out[idx];
    float xc = (mb[idx] == 0) ? 0.f : (xb[idx] - mean);
    float d = xc * w;
    acc += d*d;
  }
  red[0][tg][c] = acc;
  __syncthreads();
  if (tg == 0) {
    float s = 0.f;
#pragma unroll
    for (int g = 0; g < 8; ++g) s += red[0][g][c];
    stdsh[c] = sqrtf(s / cnt + 1e-5f);
  }
  __syncthreads();
  const float inv = 1.f / stdsh[c];
  for (int t = tg; t < TT; t += 8) {
    size_t idx = (size_t)t*CC + c;
    float xc = (mb[idx] == 0) ? 0.f : (xb[idx] - mean);
    xn_out[idx] = xc * inv;         // final xn output (overwrites wt staging)
  }
}

// ---------------------------------------------------------------------------
extern "C" void kernel_launch(void* const* d_in, const int* in_sizes, int n_in,
                              void* d_out, int out_size, void* d_ws, size_t ws_size,
                              hipStream_t stream)
{
  (void)in_sizes; (void)n_in; (void)out_size; (void)ws_size;
  const float* x        = (const float*)d_in[0];
  const int*   mask     = (const int*)  d_in[1];
  const float* xi       = (const float*)d_in[2];
  const float* tm_conv  = (const float*)d_in[3];
  const float* tm_w1    = (const float*)d_in[4];
  const float* tm_b1    = (const float*)d_in[5];
  const float* tm_w2    = (const float*)d_in[6];
  const float* tm_b2    = (const float*)d_in[7];
  const float* tm_w3    = (const float*)d_in[8];
  const float* pm_conv  = (const float*)d_in[9];
  const float* pm_w1    = (const float*)d_in[10];
  const float* pm_b1    = (const float*)d_in[11];
  const float* pm_w2    = (const float*)d_in[12];
  const float* pm_b2    = (const float*)d_in[13];
  const float* pm_w3    = (const float*)d_in[14];
  const float* rl_convx = (const float*)d_in[15];
  const float* rl_convt = (const float*)d_in[16];
  const float* rl_convp = (const float*)d_in[17];
  const float* rl_w1    = (const float*)d_in[18];
  const float* rl_b1    = (const float*)d_in[19];
  const float* rl_w2    = (const float*)d_in[20];
  const float* rl_b2    = (const float*)d_in[21];
  const float* rl_w3    = (const float*)d_in[22];
  float* out = (float*)d_out;
  float* ws  = (float*)d_ws;

  dim3 blk(32, 8);
  k1_colred       <<<BB, blk, 0, stream>>>(x, xi, tm_conv, pm_conv, rl_convx, ws);
  k2_fft_topk     <<<BB*CC, 128, 0, stream>>>(xi, ws);
  k3_mlp          <<<dim3(16, 6), 32, 0, stream>>>(ws, tm_w1, tm_b1, tm_w2, tm_b2, tm_w3,
                                                    pm_w1, pm_b1, pm_w2, pm_b2, pm_w3);
  k4_build_weights<<<BB, blk, 0, stream>>>(mask, rl_convt, rl_convp, ws, out);
  k5_ratio        <<<16, 32, 0, stream>>>(ws, rl_w1, rl_b1, rl_w2, rl_b2, rl_w3);
  k6_final        <<<BB, blk, 0, stream>>>(x, mask, ws, out);
}